// HeteroGNN_88364657147989
// MI455X (gfx1250) — compile-verified
//
#include <hip/hip_runtime.h>
#include <hip/hip_bf16.h>

typedef __attribute__((ext_vector_type(2))) float v2f;
typedef __attribute__((ext_vector_type(8))) float v8f;

#define HDIM 128
#define KDIM 256

// ---------------------------------------------------------------- utilities

__global__ void fill_kernel(float* __restrict__ p, float v, int n) {
    int i = blockIdx.x * blockDim.x + threadIdx.x;
    if (i < n) p[i] = v;
}

__global__ void deg_kernel(const int* __restrict__ dst, int E, float* __restrict__ deg) {
    int e = blockIdx.x * blockDim.x + threadIdx.x;
    if (e < E) atomicAdd(&deg[dst[e]], 1.0f);
}

__global__ void rsqrt_kernel(float* __restrict__ p, int n) {
    int i = blockIdx.x * blockDim.x + threadIdx.x;
    if (i < n) { float d = p[i]; p[i] = (d > 0.0f) ? rsqrtf(d) : 0.0f; }
}

// h[i] *= dinv[row]; agg[i] = h[i]   (self-loop seed for GCN)
__global__ void scale_copy_kernel(float* __restrict__ h, const float* __restrict__ dinv,
                                  float* __restrict__ agg, int total) {
    int i = blockIdx.x * blockDim.x + threadIdx.x;
    if (i < total) {
        float v = h[i] * dinv[i >> 7];
        h[i] = v;
        agg[i] = v;
    }
}

// one thread per (edge, channel): agg[dst] += h[src]
__global__ void scatter_kernel(const int* __restrict__ src, const int* __restrict__ dst,
                               const float* __restrict__ h, float* __restrict__ agg, int E) {
    int t = blockIdx.x * blockDim.x + threadIdx.x;
    if (t < E * HDIM) {
        int e = t >> 7, n = t & 127;
        atomicAdd(&agg[(size_t)dst[e] * HDIM + n], h[(size_t)src[e] * HDIM + n]);
    }
}

// soft = relu(dinv[row]*agg + b)
__global__ void soft_final_kernel(const float* __restrict__ agg, const float* __restrict__ dinv,
                                  const float* __restrict__ b, float* __restrict__ out, int total) {
    int i = blockIdx.x * blockDim.x + threadIdx.x;
    if (i < total) {
        float v = dinv[i >> 7] * agg[i] + b[i & 127];
        out[i] = fmaxf(v, 0.0f);
    }
}

// ---------------------------------------------------------------- WMMA GEMM
// C[nrows x 128] = X[nrows x 256] @ W[256 x 128], fp32, V_WMMA_F32_16X16X4_F32.
// W staged once per block in LDS in per-fragment lane order (each B fragment is
// one conflict-free ds_load_b64). Wave-uniform stripe-stride loop amortizes the
// LDS fill. All 8 B frags loaded into regs before the WMMA burst; next A frag
// prefetched across the burst. mode==0: write Hout. mode==1: fused article
// epilogue -> out = relu(0.5*(dinv[m]*agg + b1 + asum/max(cnt,1) + b2 + acc)).
__global__ __launch_bounds__(256) void gemm_wmma_kernel(
    const float* __restrict__ X, const float* __restrict__ W,
    float* __restrict__ Hout, int nrows, int mode,
    const float* __restrict__ agg, const float* __restrict__ bias1,
    const float* __restrict__ asum, const float* __restrict__ bias2,
    const float* __restrict__ cnt, const float* __restrict__ dinv,
    float* __restrict__ out)
{
    extern __shared__ float ldsW[];  // 256*128 floats = 128 KB, fragment order

    const int tid = threadIdx.x;
    // W[k][n] (row-major) -> fragment (k/4, n/16), lane = (n&15)+((k&2)<<3), vgpr = k&1
    for (int e = tid; e < KDIM * HDIM; e += blockDim.x) {
        int k = e >> 7, n = e & 127;
        int frag = ((k >> 2) << 3) + (n >> 4);
        int lane = (n & 15) + ((k & 2) << 3);
        int vg = k & 1;
        ldsW[frag * 64 + lane * 2 + vg] = W[e];
    }
    __syncthreads();

    const int lane = tid & 31;
    const int wave = tid >> 5;
    const int nstripes = (nrows + 15) >> 4;
    const int sstride = gridDim.x * 8;
    const int koff = (lane >> 4) << 1;            // lanes 16-31 hold K+2,K+3

    // wave-uniform stripe loop: EXEC stays all-ones for the WMMA region
    for (int stripe = blockIdx.x * 8 + wave; stripe < nstripes; stripe += sstride) {
        const int rowbase = stripe << 4;
        int row = rowbase + (lane & 15);
        if (row >= nrows) row = nrows - 1;        // clamp; masked on store
        const float* xptr = X + (size_t)row * KDIM + koff;

        v8f acc[8] = {};

        v2f a = *(const v2f*)(xptr);              // A frag: 16x4 f32
        for (int kg = 0; kg < 64; ++kg) {
            v2f bf[8];
#pragma unroll
            for (int nt = 0; nt < 8; ++nt)        // 4x ds_load_2addr_b64, one wait
                bf[nt] = *(const v2f*)(ldsW + ((kg << 3) + nt) * 64 + lane * 2);
            v2f anext = a;
            if (kg < 63) anext = *(const v2f*)(xptr + ((kg + 1) << 2));
#pragma unroll
            for (int nt = 0; nt < 8; ++nt)
                acc[nt] = __builtin_amdgcn_wmma_f32_16x16x4_f32(
                    false, a, false, bf[nt], (short)0, acc[nt], false, false);
            a = anext;
        }

        // C/D layout: lanes 0-15 -> N=lane, M=r ; lanes 16-31 -> N=lane-16, M=8+r
        const int ncol = lane & 15;
        const int mbase = (lane >> 4) << 3;

        if (mode == 0) {
#pragma unroll
            for (int r = 0; r < 8; ++r) {
                int mm = rowbase + mbase + r;
                if (mm < nrows) {
#pragma unroll
                    for (int nt = 0; nt < 8; ++nt)
                        Hout[(size_t)mm * HDIM + nt * 16 + ncol] = acc[nt][r];
                }
            }
        } else {
#pragma unroll
            for (int r = 0; r < 8; ++r) {
                int mm = rowbase + mbase + r;
                if (mm < nrows) {
                    float di = dinv[mm];
                    float rc = 1.0f / fmaxf(cnt[mm], 1.0f);
#pragma unroll
                    for (int nt = 0; nt < 8; ++nt) {
                        int col = nt * 16 + ncol;
                        size_t idx = (size_t)mm * HDIM + col;
                        float v = 0.5f * (di * agg[idx] + bias1[col]
                                          + asum[idx] * rc + bias2[col] + acc[nt][r]);
                        out[idx] = fmaxf(v, 0.0f);
                    }
                }
            }
        }
    }
}

// ---------------------------------------------------------------- launch

extern "C" void kernel_launch(void* const* d_in, const int* in_sizes, int n_in,
                              void* d_out, int out_size, void* d_ws, size_t ws_size,
                              hipStream_t stream) {
    const float* x_a   = (const float*)d_in[0];
    const float* x_s   = (const float*)d_in[1];
    const int*   ei_aa = (const int*)d_in[2];
    const int*   ei_ss = (const int*)d_in[3];
    const int*   sa_sr = (const int*)d_in[4];
    const int*   sa_ds = (const int*)d_in[5];
    const float* W_aa  = (const float*)d_in[6];
    const float* b_aa  = (const float*)d_in[7];
    const float* W_ss  = (const float*)d_in[8];
    const float* b_ss  = (const float*)d_in[9];
    const float* W_sal = (const float*)d_in[10];
    const float* b_sal = (const float*)d_in[11];
    const float* W_sar = (const float*)d_in[12];

    const int NA  = in_sizes[0] / KDIM;
    const int NS  = in_sizes[1] / KDIM;
    const int EAA = in_sizes[2] / 2;
    const int ESS = in_sizes[3] / 2;
    const int ESA = in_sizes[4];

    float* ws    = (float*)d_ws;
    float* h_a   = ws;                               // NA*128
    float* agg_a = h_a   + (size_t)NA * HDIM;        // NA*128
    float* asum  = agg_a + (size_t)NA * HDIM;        // NA*128 (SAGE sum, H space)
    float* h_s   = asum  + (size_t)NA * HDIM;        // NS*128
    float* agg_s = h_s   + (size_t)NS * HDIM;        // NS*128
    float* hsl   = agg_s + (size_t)NS * HDIM;        // NS*128 (x_s @ W_sa_l)
    float* deg_a = hsl   + (size_t)NS * HDIM;        // NA (becomes dinv)
    float* deg_s = deg_a + NA;                       // NS (becomes dinv)
    float* cnt_a = deg_s + NS;                       // NA (SAGE neighbor count)

    float* out_art  = (float*)d_out;
    float* out_soft = out_art + (size_t)NA * HDIM;

    const size_t LDSB = (size_t)KDIM * HDIM * sizeof(float);  // 128 KB
    (void)hipFuncSetAttribute((const void*)gemm_wmma_kernel,
                              hipFuncAttributeMaxDynamicSharedMemorySize, (int)LDSB);

    hipMemsetAsync(asum, 0, (size_t)NA * HDIM * sizeof(float), stream);
    hipMemsetAsync(cnt_a, 0, (size_t)NA * sizeof(float), stream);

    const int TB = 256;
    fill_kernel<<<(NA + TB - 1) / TB, TB, 0, stream>>>(deg_a, 1.0f, NA);   // self-loop
    fill_kernel<<<(NS + TB - 1) / TB, TB, 0, stream>>>(deg_s, 1.0f, NS);
    deg_kernel<<<(EAA + TB - 1) / TB, TB, 0, stream>>>(ei_aa + EAA, EAA, deg_a);
    deg_kernel<<<(ESS + TB - 1) / TB, TB, 0, stream>>>(ei_ss + ESS, ESS, deg_s);
    deg_kernel<<<(ESA + TB - 1) / TB, TB, 0, stream>>>(sa_ds, ESA, cnt_a);
    rsqrt_kernel<<<(NA + TB - 1) / TB, TB, 0, stream>>>(deg_a, NA);        // deg -> dinv
    rsqrt_kernel<<<(NS + TB - 1) / TB, TB, 0, stream>>>(deg_s, NS);

    // cap grid: amortize the 128KB W LDS fill over multiple stripes per block
    const int GCAP = 384;
    int ga = (((NA + 15) / 16) + 7) / 8; if (ga > GCAP) ga = GCAP;
    int gs = (((NS + 15) / 16) + 7) / 8; if (gs > GCAP) gs = GCAP;
    gemm_wmma_kernel<<<ga, 256, LDSB, stream>>>(x_a, W_aa, h_a, NA, 0,
        nullptr, nullptr, nullptr, nullptr, nullptr, nullptr, nullptr);
    gemm_wmma_kernel<<<gs, 256, LDSB, stream>>>(x_s, W_ss, h_s, NS, 0,
        nullptr, nullptr, nullptr, nullptr, nullptr, nullptr, nullptr);
    gemm_wmma_kernel<<<gs, 256, LDSB, stream>>>(x_s, W_sal, hsl, NS, 0,
        nullptr, nullptr, nullptr, nullptr, nullptr, nullptr, nullptr);

    int ta = NA * HDIM, ts = NS * HDIM;
    scale_copy_kernel<<<(ta + TB - 1) / TB, TB, 0, stream>>>(h_a, deg_a, agg_a, ta);
    scale_copy_kernel<<<(ts + TB - 1) / TB, TB, 0, stream>>>(h_s, deg_s, agg_s, ts);

    scatter_kernel<<<(EAA * HDIM + TB - 1) / TB, TB, 0, stream>>>(ei_aa, ei_aa + EAA, h_a, agg_a, EAA);
    scatter_kernel<<<(ESS * HDIM + TB - 1) / TB, TB, 0, stream>>>(ei_ss, ei_ss + ESS, h_s, agg_s, ESS);
    scatter_kernel<<<(ESA * HDIM + TB - 1) / TB, TB, 0, stream>>>(sa_sr, sa_ds, hsl, asum, ESA);

    soft_final_kernel<<<(ts + TB - 1) / TB, TB, 0, stream>>>(agg_s, deg_s, b_ss, out_soft, ts);

    // fused: x_a @ W_sa_r with article combine epilogue -> out_art
    gemm_wmma_kernel<<<ga, 256, LDSB, stream>>>(x_a, W_sar, nullptr, NA, 1,
        agg_a, b_aa, asum, b_sal, cnt_a, deg_a, out_art);
}